// TransformerBlock_85684597555522
// MI455X (gfx1250) — compile-verified
//
#include <hip/hip_runtime.h>
#include <hip/hip_bf16.h>
#include <math.h>

#define B_    8
#define N_    1024
#define D_    512
#define H_    8
#define DH_   64
#define ALPHA_ 0.2f
#define EPS_   1e-6f
#define NEG_   (-1e9f)

#define CH_    32          // K-elements per TDM chunk (tile_dim0)
#define PITCH_ 34          // LDS row pitch in floats (32 + 2 pad DWORDs)

typedef __attribute__((ext_vector_type(2))) float v2f;
typedef __attribute__((ext_vector_type(8))) float v8f;
typedef __attribute__((ext_vector_type(4))) unsigned int u32x4;
typedef __attribute__((ext_vector_type(4))) int i32x4;
typedef __attribute__((ext_vector_type(8))) int i32x8;

union V8 { v8f v; float f[8]; };

__device__ __forceinline__ v8f wmma4(v2f a, v2f b, v8f c) {
  // D = A(16x4 f32) * B(4x16 f32) + C(16x16 f32), full f32 precision
  return __builtin_amdgcn_wmma_f32_16x16x4_f32(false, a, false, b, (short)0, c,
                                               false, false);
}

__device__ __forceinline__ unsigned int lds_off(const void* p) {
  // low 32 bits of a generic pointer into LDS = LDS byte address
  return (unsigned int)(size_t)p;
}

// TDM: async 2D tile load (tile_d1 rows x tile_d0 elems, f32) global -> LDS,
// padded 2 DWORDs per 32-DWORD row (pad_interval code 4, amount code 1) so
// rows land at PITCH_ = 34 floats (keeps 8B alignment for b64 reads).
__device__ __forceinline__ void tdm_load_2d(unsigned int ldsa, const void* gptr,
                                            unsigned int tile_d0,
                                            unsigned int tile_d1,
                                            unsigned int tensor_d0,
                                            unsigned int tensor_d1,
                                            unsigned int stride0) {
  unsigned long long ga = (unsigned long long)(size_t)gptr;
  u32x4 g0;
  g0[0] = 1u;                                        // count=1, user descriptor
  g0[1] = ldsa;                                      // lds_addr (bytes)
  g0[2] = (unsigned int)(ga & 0xffffffffu);          // global_addr[31:0]
  g0[3] = (unsigned int)((ga >> 32) & 0x01ffffffu)   // global_addr[56:32]
        | (2u << 30);                                // type = 2 ("image")
  i32x8 g1;
  g1[0] = (int)((2u << 16)                           // data_size = 2 -> 4 bytes
        | (1u << 20)                                 // pad_enable
        | (4u << 22)                                 // pad_interval: 32 DWORDs
        | (1u << 25));                               // pad_amount: 2 DWORDs
  g1[1] = (int)(tensor_d0 << 16);                    // tensor_dim0[15:0]
  g1[2] = (int)(tensor_d1 << 16);                    // tensor_dim1[15:0]
  g1[3] = (int)(tile_d0 << 16);                      // tile_dim0
  g1[4] = (int)tile_d1;                              // tile_dim1 (tile_dim2=0)
  g1[5] = (int)stride0;                              // tensor_dim0_stride[31:0]
  g1[6] = 0;
  g1[7] = 0;
  i32x4 z4 = {0, 0, 0, 0};
#if defined(__clang_major__) && (__clang_major__ >= 23)
  i32x8 z8 = {0, 0, 0, 0, 0, 0, 0, 0};
  __builtin_amdgcn_tensor_load_to_lds(g0, g1, z4, z4, z8, 0);
#else
  __builtin_amdgcn_tensor_load_to_lds(g0, g1, z4, z4, 0);
#endif
}

// -------------------------------------------------------------------------
// Kernel 0: W[h][k][c] -> W_t[h][c][k]   (1 MB, one-off; coalesced writes)
// -------------------------------------------------------------------------
__global__ __launch_bounds__(256) void wtr_kernel(const float* __restrict__ W,
                                                  float* __restrict__ Wt) {
  int idx = blockIdx.x * 256 + threadIdx.x;   // [0, H*DH*D)
  int k  = idx & (D_ - 1);
  int c  = (idx >> 9) & (DH_ - 1);
  int hh = idx >> 15;
  Wt[idx] = W[((size_t)(hh * D_ + k)) * DH_ + c];
}

// -------------------------------------------------------------------------
// Kernel 1: LayerNorm(x) then h_t[b,h,c,n] = (xn @ W[h])^T  (WMMA f32)
// 16 rows of (b,n) per workgroup; 8 waves = 8 heads; W_t panels via TDM.
// -------------------------------------------------------------------------
__global__ __launch_bounds__(256) void ln_proj_kernel(
    const float* __restrict__ x, const float* __restrict__ g,
    const float* __restrict__ bt, const float* __restrict__ Wt,
    float* __restrict__ hout) {
  extern __shared__ char smem[];
  float* xn    = (float*)smem;              // [16][516]
  float* ssum  = xn + 16 * 516;             // [16]
  float* ssum2 = ssum + 16;                 // [16]
  float* wS    = ssum2 + 16;                // [8 heads][2 bufs][DH_*PITCH_]

  const int t  = threadIdx.x;
  const int b  = blockIdx.x >> 6;
  const int n0 = (blockIdx.x & 63) << 4;

  if (t < 16) { ssum[t] = 0.f; ssum2[t] = 0.f; }
  __syncthreads();
  {
    const int row = t >> 4, li = t & 15;
    const float* xr = x + ((size_t)(b * N_ + n0 + row)) * D_;
    float s = 0.f, s2 = 0.f;
#pragma unroll 8
    for (int j = 0; j < 32; ++j) {
      float v = xr[li + j * 16];
      xn[row * 516 + li + j * 16] = v;
      s += v; s2 += v * v;
    }
    atomicAdd(&ssum[row], s);
    atomicAdd(&ssum2[row], s2);
  }
  __syncthreads();
  {
    const int row = t >> 4, li = t & 15;
    float mean = ssum[row] * (1.0f / D_);
    float var  = ssum2[row] * (1.0f / D_) - mean * mean;
    var = var > 0.f ? var : 0.f;
    float inv = 1.0f / (sqrtf(var) + EPS_);
#pragma unroll 8
    for (int j = 0; j < 32; ++j) {
      int d = li + j * 16;
      float v = (xn[row * 516 + d] - mean) * inv;
      xn[row * 516 + d] = g[d] * v + bt[d];
    }
  }
  __syncthreads();

  const int wave = t >> 5;                 // head
  const int l    = t & 31;
  const int hi   = l >> 4;
  const int row  = l & 15;
  const float* Wh = Wt + (size_t)wave * DH_ * D_;     // [c][k], k contiguous
  float*       wB = wS + (size_t)wave * 2 * (DH_ * PITCH_);

  V8 acc[4];
#pragma unroll
  for (int q = 0; q < 4; ++q)
#pragma unroll
    for (int v = 0; v < 8; ++v) acc[q].f[v] = 0.f;

  tdm_load_2d(lds_off(wB), Wh, CH_, DH_, D_, DH_, D_);
  const int NCH = D_ / CH_;
  for (int c = 0; c < NCH; ++c) {
    if (c + 1 < NCH) {
      tdm_load_2d(lds_off(wB + ((c + 1) & 1) * (DH_ * PITCH_)),
                  Wh + (size_t)(c + 1) * CH_, CH_, DH_, D_, DH_, D_);
      __builtin_amdgcn_s_wait_tensorcnt(1);
    } else {
      __builtin_amdgcn_s_wait_tensorcnt(0);
    }
    const float* wc = wB + (c & 1) * (DH_ * PITCH_);
    for (int kk = 0; kk < CH_; kk += 4) {
      const int k0 = c * CH_ + kk;
      v2f a = *(const v2f*)(xn + row * 516 + k0 + 2 * hi);  // A[m=row][k]
#pragma unroll
      for (int q = 0; q < 4; ++q) {
        // B[k][n=q*16+row] = W_t[n][k]; (k,k+1) adjacent -> single b64 load
        v2f bb = *(const v2f*)(wc + (q * 16 + row) * PITCH_ + kk + 2 * hi);
        acc[q].v = wmma4(a, bb, acc[q].v);
      }
    }
  }

  // store transposed: h_t[b][head][c][n]
  float* hd = hout + (size_t)(b * H_ + wave) * DH_ * N_;
#pragma unroll
  for (int q = 0; q < 4; ++q)
#pragma unroll
    for (int v = 0; v < 8; ++v)
      hd[(size_t)(q * 16 + row) * N_ + n0 + v + 8 * hi] = acc[q].f[v];
}

// -------------------------------------------------------------------------
// Kernel 2: f_src/f_dst = h . a_src / a_dst  (h_t layout: coalesced on m)
// -------------------------------------------------------------------------
__global__ __launch_bounds__(256) void feat_kernel(
    const float* __restrict__ ht, const float* __restrict__ a_src,
    const float* __restrict__ a_dst, float* __restrict__ fs,
    float* __restrict__ fd) {
  int idx = blockIdx.x * 256 + threadIdx.x;        // [0, B*H*N)
  int hh  = (idx >> 10) & (H_ - 1);
  const float* hp = ht + (size_t)(idx >> 10) * DH_ * N_ + (idx & (N_ - 1));
  const float* as = a_src + hh * DH_;
  const float* ad = a_dst + hh * DH_;
  float s = 0.f, d = 0.f;
#pragma unroll 8
  for (int c = 0; c < DH_; ++c) {
    float v = hp[(size_t)c * N_];
    s += v * as[c];
    d += v * ad[c];
  }
  fs[idx] = s; fd[idx] = d;
}

// -------------------------------------------------------------------------
// Kernel 3: masked-softmax row stats (max, sum) per (b,h,n).
// -------------------------------------------------------------------------
__global__ __launch_bounds__(256) void rowstats_kernel(
    const int* __restrict__ adj, const float* __restrict__ fs,
    const float* __restrict__ fd, float* __restrict__ rmax,
    float* __restrict__ rsum) {
  extern __shared__ char smem[];
  int*   adjS = (int*)smem;                     // [32][1028]
  float* fdS  = (float*)(adjS + 32 * 1028);     // [8][1024]
  const int t  = threadIdx.x;
  const int b  = blockIdx.x >> 5;
  const int n0 = (blockIdx.x & 31) << 5;

  for (int i = t; i < 32 * 1024; i += 256) {
    int r = i >> 10, m = i & 1023;
    adjS[r * 1028 + m] = adj[((size_t)(b * N_ + n0 + r) << 10) + m];
  }
  for (int i = t; i < H_ * N_; i += 256) fdS[i] = fd[b * (H_ * N_) + i];
  __syncthreads();

  const int hh = t >> 5, nl = t & 31;
  const float fsv = fs[b * (H_ * N_) + hh * N_ + n0 + nl];
  const int*   ar = adjS + nl * 1028;
  const float* fr = fdS + hh * N_;

  float mx = -3.0e38f;
  for (int m = 0; m < N_; ++m)
    if (ar[m] > 0) {
      float v = fsv + fr[m];
      v = v > 0.f ? v : ALPHA_ * v;
      mx = fmaxf(mx, v);
    }
  float mo, sum;
  if (mx < -1e30f) {                 // no edges: softmax over all-NEG = uniform
    mo = NEG_; sum = (float)N_;
  } else {
    mo = mx; sum = 0.f;
    for (int m = 0; m < N_; ++m)
      if (ar[m] > 0) {
        float v = fsv + fr[m];
        v = v > 0.f ? v : ALPHA_ * v;
        sum += __expf(v - mo);
      }
  }
  int gi = b * (H_ * N_) + hh * N_ + n0 + nl;
  rmax[gi] = mo; rsum[gi] = sum;
}

// -------------------------------------------------------------------------
// Kernel 4: out = LN2( x + elu(attn @ h) ); A-fragments (softmax probs)
// recomputed on the fly; transposed h_t panels streamed per-wave via TDM.
// -------------------------------------------------------------------------
__global__ __launch_bounds__(256) void attn_out_kernel(
    const float* __restrict__ x, const int* __restrict__ adj,
    const float* __restrict__ ht, const float* __restrict__ fs,
    const float* __restrict__ fd, const float* __restrict__ rmax,
    const float* __restrict__ rsum, const float* __restrict__ g2,
    const float* __restrict__ b2, float* __restrict__ out) {
  extern __shared__ char smem[];
  int*   adjS = (int*)smem;                      // [16][1028]
  float* fdS  = (float*)(adjS + 16 * 1028);      // [8][1024]
  float* outS = fdS + H_ * N_;                   // [16][516]
  float* fsS  = outS + 16 * 516;                 // [128]
  float* mxS  = fsS + 128;                       // [128]
  float* isS  = mxS + 128;                       // [128]
  float* ssum  = isS + 128;                      // [16]
  float* ssum2 = ssum + 16;                      // [16]
  float* hS    = ssum2 + 16;                     // [8 heads][2][DH_*PITCH_]

  const int t  = threadIdx.x;
  const int b  = blockIdx.x >> 6;
  const int n0 = (blockIdx.x & 63) << 4;

  if (t < 16) { ssum[t] = 0.f; ssum2[t] = 0.f; }
  for (int i = t; i < 16 * 1024; i += 256) {
    int r = i >> 10, m = i & 1023;
    adjS[r * 1028 + m] = adj[((size_t)(b * N_ + n0 + r) << 10) + m];
  }
  for (int i = t; i < H_ * N_; i += 256) fdS[i] = fd[b * (H_ * N_) + i];
  if (t < 128) {
    int gi = b * (H_ * N_) + (t >> 4) * N_ + n0 + (t & 15);
    fsS[t] = fs[gi]; mxS[t] = rmax[gi]; isS[t] = 1.0f / rsum[gi];
  }
  __syncthreads();

  const int wave = t >> 5;                 // head
  const int l    = t & 31;
  const int hi   = l >> 4;
  const int row  = l & 15;
  const float frow = fsS[wave * 16 + row];
  const float mrow = mxS[wave * 16 + row];
  const float irow = isS[wave * 16 + row];
  const bool  uni  = mrow < -5e8f;
  const int*   ar  = adjS + row * 1028;
  const float* fr  = fdS + wave * N_;
  const float* hg  = ht + (size_t)(b * H_ + wave) * DH_ * N_;  // [c][m]
  float*       hB  = hS + (size_t)wave * 2 * (DH_ * PITCH_);

  V8 acc[4];
#pragma unroll
  for (int q = 0; q < 4; ++q)
#pragma unroll
    for (int v = 0; v < 8; ++v) acc[q].f[v] = 0.f;

  tdm_load_2d(lds_off(hB), hg, CH_, DH_, N_, DH_, N_);
  const int NCH = N_ / CH_;
  for (int c = 0; c < NCH; ++c) {
    if (c + 1 < NCH) {
      tdm_load_2d(lds_off(hB + ((c + 1) & 1) * (DH_ * PITCH_)),
                  hg + (size_t)(c + 1) * CH_, CH_, DH_, N_, DH_, N_);
      __builtin_amdgcn_s_wait_tensorcnt(1);
    } else {
      __builtin_amdgcn_s_wait_tensorcnt(0);
    }
    const float* hc = hB + (c & 1) * (DH_ * PITCH_);
    for (int kk = 0; kk < CH_; kk += 4) {
      const int m0 = c * CH_ + kk + 2 * hi;
      v2f a;
      {
        float p0;
        if (uni) p0 = irow;
        else if (ar[m0] > 0) {
          float v = frow + fr[m0]; v = v > 0.f ? v : ALPHA_ * v;
          p0 = __expf(v - mrow) * irow;
        } else p0 = 0.f;
        float p1;
        if (uni) p1 = irow;
        else if (ar[m0 + 1] > 0) {
          float v = frow + fr[m0 + 1]; v = v > 0.f ? v : ALPHA_ * v;
          p1 = __expf(v - mrow) * irow;
        } else p1 = 0.f;
        a.x = p0; a.y = p1;
      }
#pragma unroll
      for (int q = 0; q < 4; ++q) {
        // B[k=m][n=q*16+row] = h_t[n][m]; (m,m+1) adjacent -> single b64 load
        v2f bb = *(const v2f*)(hc + (q * 16 + row) * PITCH_ + kk + 2 * hi);
        acc[q].v = wmma4(a, bb, acc[q].v);
      }
    }
  }

  // elu + residual into shared tile
#pragma unroll
  for (int q = 0; q < 4; ++q)
#pragma unroll
    for (int v = 0; v < 8; ++v) {
      int r  = v + 8 * hi;
      int cg = wave * DH_ + q * 16 + row;
      float val = acc[q].f[v];
      val = val > 0.f ? val : (__expf(val) - 1.0f);
      val += x[((size_t)(b * N_ + n0 + r)) * D_ + cg];
      outS[r * 516 + cg] = val;
    }
  __syncthreads();

  // LN2
  const int r2 = t >> 4, li = t & 15;
  {
    float s = 0.f, s2 = 0.f;
#pragma unroll 8
    for (int j = 0; j < 32; ++j) {
      float v = outS[r2 * 516 + li + j * 16];
      s += v; s2 += v * v;
    }
    atomicAdd(&ssum[r2], s);
    atomicAdd(&ssum2[r2], s2);
  }
  __syncthreads();
  {
    float mean = ssum[r2] * (1.0f / D_);
    float var  = ssum2[r2] * (1.0f / D_) - mean * mean;
    var = var > 0.f ? var : 0.f;
    float inv = 1.0f / (sqrtf(var) + EPS_);
#pragma unroll 8
    for (int j = 0; j < 32; ++j) {
      int d = li + j * 16;
      float v = (outS[r2 * 516 + d] - mean) * inv;
      out[((size_t)(b * N_ + n0 + r2)) * D_ + d] = g2[d] * v + b2[d];
    }
  }
}

// -------------------------------------------------------------------------
extern "C" void kernel_launch(void* const* d_in, const int* in_sizes, int n_in,
                              void* d_out, int out_size, void* d_ws, size_t ws_size,
                              hipStream_t stream) {
  const float* x     = (const float*)d_in[0];
  // d_in[1] = mask (unused by reference body)
  const int*   adj   = (const int*)d_in[2];
  const float* W     = (const float*)d_in[3];
  const float* a_src = (const float*)d_in[4];
  const float* a_dst = (const float*)d_in[5];
  const float* ln1_g = (const float*)d_in[6];
  const float* ln1_b = (const float*)d_in[7];
  const float* ln2_g = (const float*)d_in[8];
  const float* ln2_b = (const float*)d_in[9];
  float* out = (float*)d_out;

  char* ws = (char*)d_ws;
  float* ht = (float*)ws;                                   // 16 MB [B][H][DH][N]
  float* fs = (float*)(ws + (size_t)B_ * H_ * DH_ * N_ * 4);
  float* fd   = fs + B_ * H_ * N_;
  float* rmax = fd + B_ * H_ * N_;
  float* rsum = rmax + B_ * H_ * N_;
  float* Wt   = rsum + B_ * H_ * N_;                        // 1 MB [H][DH][D]

  const size_t SH1 = (size_t)(16 * 516 + 32 + H_ * 2 * DH_ * PITCH_) * 4;      // ~172 KB
  const size_t SH2 = (size_t)(32 * 1028) * 4 + (size_t)(H_ * N_) * 4;          // ~164 KB
  const size_t SH3 = ((size_t)(16 * 1028) + (size_t)(H_ * N_) +
                      (size_t)(16 * 516) + (3 * 128 + 32) +
                      (size_t)(H_ * 2 * DH_ * PITCH_)) * 4;                    // ~272 KB

  wtr_kernel<<<(H_ * DH_ * D_) / 256, 256, 0, stream>>>(W, Wt);
  ln_proj_kernel<<<B_ * (N_ / 16), 256, SH1, stream>>>(x, ln1_g, ln1_b, Wt, ht);
  feat_kernel<<<(B_ * H_ * N_) / 256, 256, 0, stream>>>(ht, a_src, a_dst, fs, fd);
  rowstats_kernel<<<B_ * (N_ / 32), 256, SH2, stream>>>(adj, fs, fd, rmax, rsum);
  attn_out_kernel<<<B_ * (N_ / 16), 256, SH3, stream>>>(x, adj, ht, fs, fd, rmax,
                                                        rsum, ln2_g, ln2_b, out);
}